// VectorQuantizer_18519898980940
// MI455X (gfx1250) — compile-verified
//
#include <hip/hip_runtime.h>

#define D 512
#define K_CODES 8192
#define N_ROWS 16384
#define TILE_M 128
#define TILE_N 64
#define KC 64
#define A_STRIDE 516            // floats; 4*r mod 64 -> conflict-free, 16B-aligned rows
#define B_STRIDE 68             // floats; same property
#define A_LDS_FLOATS (TILE_M * A_STRIDE)   // 66048
#define B_LDS_FLOATS (TILE_N * B_STRIDE)   // 4352 (per buffer; double-buffered)
#define COL_SPLITS 4
#define CT_PER_SPLIT ((K_CODES / TILE_N) / COL_SPLITS)  // 32
#define N_CHUNK (D / KC)                                // 8
#define TOTAL_Q (CT_PER_SPLIT * N_CHUNK)                // 256

typedef __attribute__((ext_vector_type(2))) float v2f;
typedef __attribute__((ext_vector_type(4))) float v4f;
typedef __attribute__((ext_vector_type(8))) float v8f;

// ---------------- kernel 1: ||e||^2 per code row (one wave per row) ----------------
__global__ void __launch_bounds__(256) enorm_kernel(const float* __restrict__ E,
                                                    float* __restrict__ enorm) {
  const int gid  = blockIdx.x * blockDim.x + threadIdx.x;
  const int wave = gid >> 5;
  const int lane = threadIdx.x & 31;
  const float* row = E + (size_t)wave * D;
  float s = 0.f;
#pragma unroll
  for (int j = 0; j < D; j += 128) {
    v4f v = *(const v4f*)(row + j + lane * 4);
    s += v.x * v.x + v.y * v.y + v.z * v.z + v.w * v.w;
  }
#pragma unroll
  for (int m = 16; m >= 1; m >>= 1) s += __shfl_xor(s, m, 32);
  if (lane == 0) enorm[wave] = s;
}

// ---------------- kernel 2: fp32 WMMA distance GEMM + running argmin ----------------
// grid = 128 row-tiles * 4 col-splits; block = 256 threads (8 waves).
// A (128x512 x-tile) resident in LDS; B double-buffered (integer-offset selected
// so the DS address space is preserved), software pipelined.
__global__ void __launch_bounds__(256, 1)
argmin_kernel(const float* __restrict__ X, const float* __restrict__ E,
              const float* __restrict__ enorm,
              float* __restrict__ candD, int* __restrict__ candI) {
  extern __shared__ float lds[];

  const int t    = threadIdx.x;
  const int lane = t & 31;
  const int w    = t >> 5;
  const int hi   = lane >> 4;            // 0: C rows r..r+7 ; 1: rows r+8..r+15
  const int l16  = lane & 15;
  const int rt   = blockIdx.x & 127;
  const int cs   = blockIdx.x >> 7;
  const int rowBase = rt * TILE_M;
  const int splitColBase = cs * (CT_PER_SPLIT * TILE_N);  // cs * 2048

  // Stage the full 128x512 x-tile into LDS once (resident for all col tiles).
#pragma unroll 4
  for (int i = 0; i < 64; ++i) {
    int idx = t + 256 * i;
    int r   = idx >> 7;                  // 128 float4 per row
    int c4  = idx & 127;
    v4f v = *(const v4f*)(X + (size_t)(rowBase + r) * D + c4 * 4);
    *(v4f*)(lds + r * A_STRIDE + c4 * 4) = v;
  }

  // B staging geometry: 1024 float4 per chunk / 256 threads = 4 each.
  int rB[4], cB[4];
#pragma unroll
  for (int i = 0; i < 4; ++i) {
    int idx = t + 256 * i;
    rB[i] = idx >> 4;                    // code row within tile
    cB[i] = (idx & 15) * 4;              // float offset within 64-float chunk
  }

  v4f breg[4];
  auto issueLoads = [&](int q) {
    const int colBase = splitColBase + (q >> 3) * TILE_N;
    const int dcf     = (q & 7) * KC;
#pragma unroll
    for (int i = 0; i < 4; ++i)
      breg[i] = *(const v4f*)(E + (size_t)(colBase + rB[i]) * D + dcf + cB[i]);
  };
  // store into LDS buffer selected by integer offset (keeps DS address space)
  auto storeLds = [&](int bufOff) {
#pragma unroll
    for (int i = 0; i < 4; ++i)
      *(v4f*)(lds + bufOff + rB[i] * B_STRIDE + cB[i]) = breg[i];
  };

  float best[8];
  int   bidx[8];
#pragma unroll
  for (int r = 0; r < 8; ++r) { best[r] = 3.4028235e38f; bidx[r] = 0; }

  // A frag: lane -> row (w*16 + l16), K-pair offset hi*2. B frag mirrored.
  const int aOff = (w * 16 + l16) * A_STRIDE + hi * 2;
  const int bOff = l16 * B_STRIDE + hi * 2;

  // ---- software pipeline prologue ----
  issueLoads(0);
  storeLds(A_LDS_FLOATS);

  for (int ct = 0; ct < CT_PER_SPLIT; ++ct) {
    const int colBase = splitColBase + ct * TILE_N;

    v8f acc[4];
#pragma unroll
    for (int c = 0; c < 4; ++c) acc[c] = 0.0f;   // straight-line reset (no predication)

    for (int dc = 0; dc < N_CHUNK; ++dc) {
      const int q = ct * N_CHUNK + dc;
      const int curOff  = A_LDS_FLOATS + (q & 1) * B_LDS_FLOATS;
      const int nextOff = A_LDS_FLOATS + ((q & 1) ^ 1) * B_LDS_FLOATS;
      if (q + 1 < TOTAL_Q) issueLoads(q + 1);    // overlap L2 latency with compute

      __syncthreads();  // publishes current-buffer stores; guards 2-iter buffer reuse

      const float* aa = lds + aOff + dc * KC;
      const float* bb = lds + curOff + bOff;
#pragma unroll
      for (int k = 0; k < KC; k += 4) {
        v2f a  = *(const v2f*)(aa + k);
        v2f b0 = *(const v2f*)(bb + 0 * 16 * B_STRIDE + k);
        v2f b1 = *(const v2f*)(bb + 1 * 16 * B_STRIDE + k);
        v2f b2 = *(const v2f*)(bb + 2 * 16 * B_STRIDE + k);
        v2f b3 = *(const v2f*)(bb + 3 * 16 * B_STRIDE + k);
        acc[0] = __builtin_amdgcn_wmma_f32_16x16x4_f32(false, a, false, b0, (short)0, acc[0], false, false);
        acc[1] = __builtin_amdgcn_wmma_f32_16x16x4_f32(false, a, false, b1, (short)0, acc[1], false, false);
        acc[2] = __builtin_amdgcn_wmma_f32_16x16x4_f32(false, a, false, b2, (short)0, acc[2], false, false);
        acc[3] = __builtin_amdgcn_wmma_f32_16x16x4_f32(false, a, false, b3, (short)0, acc[3], false, false);
      }

      if (q + 1 < TOTAL_Q) storeLds(nextOff);    // fill other buffer
    }

    // Fold distances d = ||e||^2 - 2*sim into running per-lane argmin.
    // C layout: lane -> col = colBase + 16c + l16 ; VGPR r -> row = 16w + 8*hi + r.
#pragma unroll
    for (int c = 0; c < 4; ++c) {
      int col = colBase + c * 16 + l16;
      float s2 = enorm[col];
#pragma unroll
      for (int r = 0; r < 8; ++r) {
        float d = fmaf(-2.0f, acc[c][r], s2);
        if (d < best[r]) { best[r] = d; bidx[r] = col; }  // scan order => first-min ties
      }
    }
  }

  // Cross-lane min+index over the 16-lane group holding the same row.
#pragma unroll
  for (int r = 0; r < 8; ++r) {
    float d = best[r];
    int   ix = bidx[r];
#pragma unroll
    for (int m = 1; m < 16; m <<= 1) {
      float od = __shfl_xor(d, m, 32);
      int   oi = __shfl_xor(ix, m, 32);
      if (od < d || (od == d && oi < ix)) { d = od; ix = oi; }
    }
    if (l16 == 0) {
      int row = rowBase + w * 16 + hi * 8 + r;
      candD[row * COL_SPLITS + cs] = d;
      candI[row * COL_SPLITS + cs] = ix;
    }
  }
}

// ---------------- kernel 3: combine splits, gather, outputs, loss partials ----------------
__global__ void __launch_bounds__(256)
gather_kernel(const float* __restrict__ X, const float* __restrict__ E,
              const float* __restrict__ candD, const int* __restrict__ candI,
              float* __restrict__ out, float* __restrict__ quant,
              float* __restrict__ partials) {
  __shared__ float wsum[8];
  const int t = threadIdx.x, lane = t & 31, w = t >> 5;
  const int row = blockIdx.x * 8 + w;

  float bd = candD[row * COL_SPLITS];
  int   bi = candI[row * COL_SPLITS];
#pragma unroll
  for (int s = 1; s < COL_SPLITS; ++s) {
    float d2 = candD[row * COL_SPLITS + s];
    int   i2 = candI[row * COL_SPLITS + s];
    if (d2 < bd || (d2 == bd && i2 < bi)) { bd = d2; bi = i2; }
  }

  const float* xr = X + (size_t)row * D;
  const float* er = E + (size_t)bi * D;
  float* o1 = out   + (size_t)row * D;
  float* o2 = quant + (size_t)row * D;
  float s = 0.f;
#pragma unroll
  for (int j = 0; j < D; j += 128) {
    int off = j + lane * 4;
    v4f q  = *(const v4f*)(er + off);
    v4f xv = *(const v4f*)(xr + off);
    *(v4f*)(o1 + off) = q;                 // out = x + sg(q - x) == q
    *(v4f*)(o2 + off) = q;                 // quantized
    v4f dd = q - xv;
    s += dd.x * dd.x + dd.y * dd.y + dd.z * dd.z + dd.w * dd.w;
  }
#pragma unroll
  for (int m = 16; m >= 1; m >>= 1) s += __shfl_xor(s, m, 32);
  if (lane == 0) wsum[w] = s;
  __syncthreads();
  if (t == 0) {
    float b = 0.f;
#pragma unroll
    for (int i = 0; i < 8; ++i) b += wsum[i];
    partials[blockIdx.x] = b;              // deterministic per-block partial
  }
}

// ---------------- kernel 4: deterministic final loss reduction ----------------
__global__ void __launch_bounds__(256)
loss_kernel(const float* __restrict__ partials, float* __restrict__ lossOut) {
  __shared__ float sm[256];
  const int t = threadIdx.x;
  float s = 0.f;
  for (int i = t; i < 2048; i += 256) s += partials[i];
  sm[t] = s;
  __syncthreads();
  for (int m = 128; m >= 1; m >>= 1) {
    if (t < m) sm[t] += sm[t + m];
    __syncthreads();
  }
  // loss = (BETA + 1) * mean((q-x)^2) = 1.25 * sum / (16384*512)
  if (t == 0) lossOut[0] = sm[0] * (1.25f / (16384.0f * 512.0f));
}

extern "C" void kernel_launch(void* const* d_in, const int* in_sizes, int n_in,
                              void* d_out, int out_size, void* d_ws, size_t ws_size,
                              hipStream_t stream) {
  const float* X = (const float*)d_in[0];      // [16,1024,512] fp32
  const float* E = (const float*)d_in[1];      // [8192,512] fp32

  float* out   = (float*)d_out;                            // 16384*512
  float* quant = out + (size_t)N_ROWS * D;                 // 16384*512
  float* loss  = out + (size_t)2 * N_ROWS * D;             // 1

  // workspace layout
  float* enorm    = (float*)d_ws;                          // 8192 f
  float* candD    = enorm + K_CODES;                       // 16384*4 f
  int*   candI    = (int*)(candD + (size_t)N_ROWS * COL_SPLITS); // 16384*4 i
  float* partials = (float*)(candI + (size_t)N_ROWS * COL_SPLITS); // 2048 f

  enorm_kernel<<<K_CODES / 8, 256, 0, stream>>>(E, enorm);

  const size_t ldsBytes = (size_t)(A_LDS_FLOATS + 2 * B_LDS_FLOATS) * sizeof(float); // 299008
  argmin_kernel<<<(N_ROWS / TILE_M) * COL_SPLITS, 256, ldsBytes, stream>>>(
      X, E, enorm, candD, candI);

  gather_kernel<<<N_ROWS / 8, 256, 0, stream>>>(X, E, candD, candI, out, quant, partials);

  loss_kernel<<<1, 256, 0, stream>>>(partials, loss);
}